// GCNEvaluator_61598420959301
// MI455X (gfx1250) — compile-verified
//
#include <hip/hip_runtime.h>
#include <math.h>

// ---------------------------------------------------------------------------
// GCN evaluator for MI455X (gfx1250, wave32).
// Dense math: V_WMMA_F32_16X16X4_F32 (exact f32, K=4 chained).
// Weights are pre-packed once per launch into zero-padded TRANSPOSED 16x16
// tables so every WMMA B-operand chunk is one unconditional b64 load.
// Edge weights stored compact [E][10] so the dominant per-layer stream is
// fully coalesced with no pad bytes.
// ---------------------------------------------------------------------------

typedef float v2f __attribute__((ext_vector_type(2)));
typedef float v8f __attribute__((ext_vector_type(8)));

#define HID 10
#define HP  16
#define WPB 8   // waves per 256-thread block

__device__ __forceinline__ v8f wmma4(v2f a, v2f b, v8f c) {
  // (neg_a, A, neg_b, B, c_mod, C, reuse_a, reuse_b)
  return __builtin_amdgcn_wmma_f32_16x16x4_f32(false, a, false, b, (short)0, c,
                                               false, false);
}

// B chunk from a packed transposed 16x16 table: Wt[n][k] = W[k][n] (0-padded).
// lane L needs B[4*kchunk + 2*(L>>4) + v][L&15]  ->  contiguous pair in Wt.
__device__ __forceinline__ v2f load_Bt(const float* __restrict__ Wt, int kchunk,
                                       int lane) {
  int n = lane & 15;
  int k = 4 * kchunk + 2 * (lane >> 4);
  return *(const v2f*)(Wt + n * 16 + k);
}

// ---------------- K0: pack weights into padded transposed tables -----------
// slot 0: ew_W1 (4x10)   slot 1: ew_W2 (10x10)   slot 2: iW (16x10)
// slot 3+2l: lin_W[l][0:10]   slot 4+2l: lin_W[l][10:20]
__global__ void __launch_bounds__(256) k_pack_weights(
    const float* __restrict__ W1, const float* __restrict__ W2,
    const float* __restrict__ iW, const float* __restrict__ linW,
    float* __restrict__ wpad) {
  int i = blockIdx.x * 256 + threadIdx.x;
  int slot = blockIdx.x;
  int n = (threadIdx.x >> 4);   // original column
  int k = (threadIdx.x & 15);   // original row
  float v = 0.f;
  if (n < HID) {
    if (slot == 0) {
      if (k < 4) v = W1[k * HID + n];
    } else if (slot == 1) {
      if (k < HID) v = W2[k * HID + n];
    } else if (slot == 2) {
      v = iW[k * HID + n];                      // iW has 16 rows
    } else {
      int l = (slot - 3) >> 1, half = (slot - 3) & 1;
      if (k < HID)
        v = linW[((size_t)l * 2 * HID + half * HID + k) * HID + n];
    }
  }
  (void)i;
  wpad[(size_t)slot * 256 + (size_t)n * 16 + k] = v;
}

// ---------------- K1: deg init (self-loop weight = 1) ----------------------
__global__ void __launch_bounds__(256) k_deg_init(float* __restrict__ deg,
                                                  long long total) {
  long long i = (long long)blockIdx.x * blockDim.x + threadIdx.x;
  if (i < total) deg[i] = 1.0f;
}

// ---------------- K2: edge MLP + degree accumulation -----------------------
// ew = relu(relu(edge_attr @ W1) @ W2), stored compact [E][HID];
// fused atomicAdd into deg (L2-resident table).
__global__ void __launch_bounds__(256) k_edge_mlp(
    const float* __restrict__ edge_attr, const int* __restrict__ colIdx,
    const float* __restrict__ W1t, const float* __restrict__ W2t,
    float* __restrict__ ew, float* __restrict__ deg, int ntiles) {
  __shared__ float lds[WPB][16][18];
  int lane = threadIdx.x & 31;
  int wid  = threadIdx.x >> 5;
  int tile = blockIdx.x * WPB + wid;
  bool act = tile < ntiles;
  int t  = act ? tile : 0;              // clamp: keep EXEC full through WMMA
  int tb = t * 16;
  int n  = lane & 15;
  int hi = lane >> 4;

  // GEMM1: [16 edges x 4] @ [4 x 10]  (single K=4 WMMA)
  v2f a0 = *(const v2f*)(edge_attr + (size_t)(tb + n) * 4 + 2 * hi);
  v2f b0 = load_Bt(W1t, 0, lane);
  v8f t1 = {};
  t1 = wmma4(a0, b0, t1);

  // D-layout -> LDS (row major) for A-layout reload
#pragma unroll
  for (int v = 0; v < 8; ++v) lds[wid][v + 8 * hi][n] = fmaxf(t1[v], 0.f);
  __syncthreads();

  // GEMM2: [16 x 10] @ [10 x 10]  (3 chained K=4 WMMAs, zero-padded)
  v8f t2 = {};
#pragma unroll
  for (int c = 0; c < 3; ++c) {
    int k = 4 * c + 2 * hi;
    v2f a;
    a.x = lds[wid][n][k];
    a.y = lds[wid][n][k + 1];
    v2f b = load_Bt(W2t, c, lane);
    t2 = wmma4(a, b, t2);
  }

  if (act && n < HID) {
#pragma unroll
    for (int v = 0; v < 8; ++v) {
      int e = tb + v + 8 * hi;
      float val = fmaxf(t2[v], 0.f);
      ew[(size_t)e * HID + n] = val;              // compact store
      if (val > 0.f) {
        int ci = colIdx[e];                       // broadcast per half-wave
        atomicAdd(deg + (size_t)ci * HP + n, val);
      }
    }
  }
}

// ---------------- K3: dinv = rsqrt(deg)  (deg >= 1 always) -----------------
__global__ void __launch_bounds__(256) k_rsqrt(float* __restrict__ deg,
                                               long long total) {
  long long i = (long long)blockIdx.x * blockDim.x + threadIdx.x;
  if (i < total) deg[i] = rsqrtf(deg[i]);
}

// ---------------- K4: x_ = x @ iW + ib   (IN=16 -> 4 chained WMMAs) --------
__global__ void __launch_bounds__(256) k_node_in(
    const float* __restrict__ x, const float* __restrict__ iWt,
    const float* __restrict__ ib, float* __restrict__ xfeat, int ntiles) {
  int lane = threadIdx.x & 31;
  int wid  = threadIdx.x >> 5;
  int tile = blockIdx.x * WPB + wid;
  bool act = tile < ntiles;
  int t  = act ? tile : 0;
  int tb = t * 16;
  int n  = lane & 15;
  int hi = lane >> 4;

  v8f d = {};
#pragma unroll
  for (int c = 0; c < 4; ++c) {
    v2f a = *(const v2f*)(x + (size_t)(tb + n) * 16 + 4 * c + 2 * hi);
    v2f b = load_Bt(iWt, c, lane);
    d = wmma4(a, b, d);
  }
  float bias = (n < HID) ? ib[n] : 0.f;
  if (act) {
#pragma unroll
    for (int v = 0; v < 8; ++v)
      xfeat[(size_t)(tb + v + 8 * hi) * HP + n] = d[v] + bias;
  }
}

// ---------------- K5a: xp = relu(h)*gcn_w ; agg = dinv^2*xp + gcn_b --------
__global__ void __launch_bounds__(256) k_layer_pre(
    const float* __restrict__ hin, const float* __restrict__ dinv,
    const float* __restrict__ gw, const float* __restrict__ gb,
    float* __restrict__ xp, float* __restrict__ agg, long long total) {
  long long i = (long long)blockIdx.x * blockDim.x + threadIdx.x;
  if (i >= total) return;
  int h = (int)(i & (HP - 1));
  float w  = (h < HID) ? gw[h] : 0.f;
  float b  = (h < HID) ? gb[h] : 0.f;
  float v  = fmaxf(hin[i], 0.f) * w;
  float di = dinv[i];
  xp[i]  = v;
  agg[i] = di * di * v + b;   // self-loop term fused in
}

// ---------------- K5b: edge scatter with on-the-fly normalization ----------
// one thread per (edge, channel); ew compact [E][HID] -> coalesced stream.
__global__ void __launch_bounds__(256) k_layer_scatter(
    const float* __restrict__ ew, const int* __restrict__ rowIdx,
    const int* __restrict__ colIdx, const float* __restrict__ dinv,
    const float* __restrict__ xp, float* __restrict__ agg, long long nelem) {
  long long i = (long long)blockIdx.x * blockDim.x + threadIdx.x;
  if (i >= nelem) return;
  float w = ew[i];                       // streaming read, fully coalesced
  if (w <= 0.f) return;                  // relu zeros -> skip atomic
  long long e = i / HID;                 // constant divide -> mulhi
  int h = (int)(i - e * HID);
  int r = rowIdx[e], c = colIdx[e];
  float val = dinv[(size_t)r * HP + h] * w * dinv[(size_t)c * HP + h] *
              xp[(size_t)r * HP + h];    // gathers hit L2 (6.4MB tables)
  atomicAdd(agg + (size_t)c * HP + h, val);
}

// ---------------- K5c: h = x_ @ W_lo + agg @ W_hi + x_  (6 WMMAs) ----------
__global__ void __launch_bounds__(256) k_layer_update(
    const float* __restrict__ xfeat, const float* __restrict__ agg,
    const float* __restrict__ Wlot, const float* __restrict__ Whit,
    float* __restrict__ hout, int ntiles) {
  int lane = threadIdx.x & 31;
  int wid  = threadIdx.x >> 5;
  int tile = blockIdx.x * WPB + wid;
  bool act = tile < ntiles;
  int t  = act ? tile : 0;
  int tb = t * 16;
  int n  = lane & 15;
  int hi = lane >> 4;

  v8f d;
#pragma unroll
  for (int v = 0; v < 8; ++v)            // residual: C seeded with x_ (D-layout)
    d[v] = xfeat[(size_t)(tb + v + 8 * hi) * HP + n];
#pragma unroll
  for (int c = 0; c < 3; ++c) {          // x_ @ lin_W[l][0:10]
    v2f a = *(const v2f*)(xfeat + (size_t)(tb + n) * HP + 4 * c + 2 * hi);
    d = wmma4(a, load_Bt(Wlot, c, lane), d);
  }
#pragma unroll
  for (int c = 0; c < 3; ++c) {          // agg @ lin_W[l][10:20]
    v2f a = *(const v2f*)(agg + (size_t)(tb + n) * HP + 4 * c + 2 * hi);
    d = wmma4(a, load_Bt(Whit, c, lane), d);
  }
  if (act) {
#pragma unroll
    for (int v = 0; v < 8; ++v)
      hout[(size_t)(tb + v + 8 * hi) * HP + n] = d[v];
  }
}

// ---------------- K6: softmax column 1 = sigmoid(l1 - l0) ------------------
__global__ void __launch_bounds__(256) k_out(
    const float* __restrict__ xfeat, const float* __restrict__ hbuf,
    const float* __restrict__ oW, float* __restrict__ out, int N_) {
  int i = blockIdx.x * blockDim.x + threadIdx.x;
  if (i >= N_) return;
  float l0 = 0.f, l1 = 0.f;
#pragma unroll
  for (int k = 0; k < HID; ++k) {
    float xv = xfeat[(size_t)i * HP + k];
    l0 += xv * oW[k * 2 + 0];
    l1 += xv * oW[k * 2 + 1];
  }
#pragma unroll
  for (int k = 0; k < HID; ++k) {
    float hv = fmaxf(hbuf[(size_t)i * HP + k], 0.f);
    l0 += hv * oW[(HID + k) * 2 + 0];
    l1 += hv * oW[(HID + k) * 2 + 1];
  }
  out[i] = 1.f / (1.f + expf(l0 - l1));
}

// ---------------------------------------------------------------------------
extern "C" void kernel_launch(void* const* d_in, const int* in_sizes, int n_in,
                              void* d_out, int out_size, void* d_ws,
                              size_t ws_size, hipStream_t stream) {
  const float* x      = (const float*)d_in[0];
  const int*   eidx   = (const int*)d_in[1];
  const float* eattr  = (const float*)d_in[2];
  const float* ew_W1  = (const float*)d_in[3];
  const float* ew_W2  = (const float*)d_in[4];
  const float* iW     = (const float*)d_in[5];
  const float* ib     = (const float*)d_in[6];
  const float* gcn_w  = (const float*)d_in[7];
  const float* gcn_b  = (const float*)d_in[8];
  const float* lin_W  = (const float*)d_in[9];
  const float* oW     = (const float*)d_in[10];
  float* out = (float*)d_out;

  const int       N_ = in_sizes[0] / 16;      // IN = 16
  const long long E_ = in_sizes[1] / 2;       // edge_index [2, E]
  const int L_ = in_sizes[7] / HID;           // num layers
  const int* rowIdx = eidx;
  const int* colIdx = eidx + E_;

  // workspace layout (bytes); all offsets 8B-aligned
  char* ws = (char*)d_ws;
  float* ew    = (float*)ws;                       // compact [E][HID]
  size_t off   = (size_t)E_ * HID * sizeof(float);
  float* dinv  = (float*)(ws + off); off += (size_t)N_ * HP * sizeof(float);
  float* xfeat = (float*)(ws + off); off += (size_t)N_ * HP * sizeof(float);
  float* xp    = (float*)(ws + off); off += (size_t)N_ * HP * sizeof(float);
  float* agg   = (float*)(ws + off); off += (size_t)N_ * HP * sizeof(float);
  float* hbuf  = (float*)(ws + off); off += (size_t)N_ * HP * sizeof(float);
  float* wpad  = (float*)(ws + off);               // packed weight tables

  const long long nNodeEl = (long long)N_ * HP;
  const long long nEdgeEl = E_ * HID;
  const int etiles  = (int)(E_ / 16);
  const int ntilesN = N_ / 16;
  const int nslots  = 3 + 2 * L_;

  // 0) pack weights (zero-padded, transposed 16x16 tables)
  k_pack_weights<<<nslots, 256, 0, stream>>>(ew_W1, ew_W2, iW, lin_W, wpad);
  // 1) deg init (dinv buffer doubles as deg)
  k_deg_init<<<(unsigned)((nNodeEl + 255) / 256), 256, 0, stream>>>(dinv, nNodeEl);
  // 2) edge MLP + degree atomics
  k_edge_mlp<<<(unsigned)((etiles + WPB - 1) / WPB), 256, 0, stream>>>(
      eattr, colIdx, wpad, wpad + 256, ew, dinv, etiles);
  // 3) dinv = rsqrt(deg)
  k_rsqrt<<<(unsigned)((nNodeEl + 255) / 256), 256, 0, stream>>>(dinv, nNodeEl);
  // 4) x_ = x @ iW + ib
  k_node_in<<<(unsigned)((ntilesN + WPB - 1) / WPB), 256, 0, stream>>>(
      x, wpad + 512, ib, xfeat, ntilesN);

  const float* hin = xfeat;
  for (int l = 0; l < L_; ++l) {
    k_layer_pre<<<(unsigned)((nNodeEl + 255) / 256), 256, 0, stream>>>(
        hin, dinv, gcn_w + (size_t)l * HID, gcn_b + (size_t)l * HID, xp, agg,
        nNodeEl);
    k_layer_scatter<<<(unsigned)((nEdgeEl + 255) / 256), 256, 0, stream>>>(
        ew, rowIdx, colIdx, dinv, xp, agg, nEdgeEl);
    k_layer_update<<<(unsigned)((ntilesN + WPB - 1) / WPB), 256, 0, stream>>>(
        xfeat, agg, wpad + (size_t)(3 + 2 * l) * 256,
        wpad + (size_t)(4 + 2 * l) * 256, hbuf, ntilesN);
    hin = hbuf;
  }
  // 5) output probabilities
  k_out<<<(unsigned)((N_ + 255) / 256), 256, 0, stream>>>(xfeat, hbuf, oW, out, N_);
}